// GATEncoder_33432025432487
// MI455X (gfx1250) — compile-verified
//
#include <hip/hip_runtime.h>
#include <hip/hip_bf16.h>

typedef __attribute__((ext_vector_type(2))) float v2f;
typedef __attribute__((ext_vector_type(8))) float v8f;

#define NEG_INF (-__builtin_huge_valf())

// ---------------------------------------------------------------- init kernels
__global__ void fill_f32_kernel(float* __restrict__ p, float v, int n) {
    int i = blockIdx.x * blockDim.x + threadIdx.x;
    if (i < n) p[i] = v;
}

// p[node*ch + c] = b[c]   (seeds aggregation buffers with the bias)
__global__ void fill_bias_kernel(float* __restrict__ p, const float* __restrict__ b,
                                 int n_nodes, int ch) {
    int i = blockIdx.x * blockDim.x + threadIdx.x;
    if (i < n_nodes * ch) p[i] = b[i % ch];
}

// ---------------------------------------------------------------- WMMA GEMM
// C[nrows, NOUT] = A[nrows, K] @ W[K, NOUT]
// Block: 32*(NOUT/16) threads. blockIdx.x = 16-row M tile. One wave per 16x16 tile.
// Uses V_WMMA_F32_16X16X4_F32 (fp32 A/B, fp32 accum) -- native fp32 matrix path.
template <int K, int NOUT>
__global__ void __launch_bounds__(32 * (NOUT / 16))
gemm_wmma_kernel(const float* __restrict__ A, const float* __restrict__ W,
                 float* __restrict__ C, int nrows) {
    __shared__ float As[16][K];  // 16 rows of A staged once per block (8 KB)
    const int tid = threadIdx.x;
    const int m_base = blockIdx.x * 16;
    for (int i = tid; i < 16 * K; i += blockDim.x) {
        int r = i / K, c = i % K;
        int row = m_base + r;
        As[r][c] = (row < nrows) ? A[(size_t)row * K + c] : 0.0f;
    }
    __syncthreads();

    const int wave   = tid >> 5;
    const int lane   = tid & 31;
    const int n_base = wave * 16;
    const int m      = lane & 15;          // row within A frag / col within B frag
    const int koff   = (lane >> 4) << 1;   // lanes 16-31 hold K+2,K+3

    v8f acc = {};
    for (int k0 = 0; k0 < K; k0 += 4) {
        v2f a, b;
        // A 16x4 layout: lanes 0-15: M=lane, {K=k0, k0+1}; lanes 16-31: {k0+2, k0+3}
        a.x = As[m][k0 + koff];
        a.y = As[m][k0 + koff + 1];
        // B 4x16 layout (mirror): VGPR0 row k0 (lanes 0-15) / k0+2 (16-31), VGPR1 k0+1 / k0+3
        b.x = W[(size_t)(k0 + koff) * NOUT + n_base + m];
        b.y = W[(size_t)(k0 + koff + 1) * NOUT + n_base + m];
        acc = __builtin_amdgcn_wmma_f32_16x16x4_f32(false, a, false, b, (short)0, acc,
                                                    false, false);
    }
    // C layout: VGPR i -> M = i (lanes 0-15) or i+8 (lanes 16-31), N = lane&15
    #pragma unroll
    for (int i = 0; i < 8; ++i) {
        int row = m_base + i + ((lane >> 4) << 3);
        if (row < nrows) C[(size_t)row * NOUT + n_base + m] = acc[i];
    }
}

// ---------------------------------------------------------------- attention logits
// als[n,h] = dot(h[n,h,:], a_src[h,:]);  ald likewise.
template <int HEADS, int CH>
__global__ void att_logits_kernel(const float* __restrict__ h,
                                  const float* __restrict__ a_src,
                                  const float* __restrict__ a_dst,
                                  float* __restrict__ als, float* __restrict__ ald,
                                  int n_nodes) {
    int idx = blockIdx.x * blockDim.x + threadIdx.x;
    if (idx >= n_nodes * HEADS) return;
    int node = idx / HEADS, head = idx % HEADS;
    const float* hp = h + (size_t)node * HEADS * CH + head * CH;
    float s = 0.f, d = 0.f;
    #pragma unroll 8
    for (int c = 0; c < CH; ++c) {
        float v = hp[c];
        s += v * a_src[head * CH + c];
        d += v * a_dst[head * CH + c];
    }
    als[idx] = s;
    ald[idx] = d;
}

__device__ __forceinline__ void edge_endpoints(const long long* __restrict__ ei,
                                               int E, int eid, int& s, int& d) {
    if (eid < E) {
        s = (int)ei[eid];
        d = (int)ei[(size_t)E + eid];
    } else {
        s = d = eid - E;  // self loop
    }
}

// ---------------------------------------------------------------- edge pass 1: segment max
template <int HEADS>
__global__ void edge_max_kernel(const long long* __restrict__ ei, int E, int Etot,
                                const float* __restrict__ als,
                                const float* __restrict__ ald,
                                float* __restrict__ mx) {
    int idx = blockIdx.x * blockDim.x + threadIdx.x;
    if (idx >= Etot * HEADS) return;
    int eid = idx / HEADS, head = idx % HEADS;
    int s, d;
    edge_endpoints(ei, E, eid, s, d);
    float e = als[s * HEADS + head] + ald[d * HEADS + head];
    e = (e >= 0.f) ? e : 0.2f * e;  // leaky_relu slope 0.2
    atomicMax(&mx[d * HEADS + head], e);
}

// ---------------------------------------------------------------- edge pass 2: segment sum
template <int HEADS>
__global__ void edge_sum_kernel(const long long* __restrict__ ei, int E, int Etot,
                                const float* __restrict__ als,
                                const float* __restrict__ ald,
                                const float* __restrict__ mx,
                                float* __restrict__ den) {
    int idx = blockIdx.x * blockDim.x + threadIdx.x;
    if (idx >= Etot * HEADS) return;
    int eid = idx / HEADS, head = idx % HEADS;
    int s, d;
    edge_endpoints(ei, E, eid, s, d);
    float e = als[s * HEADS + head] + ald[d * HEADS + head];
    e = (e >= 0.f) ? e : 0.2f * e;
    atomicAdd(&den[d * HEADS + head], __expf(e - mx[d * HEADS + head]));
}

// ---------------------------------------------------------------- edge pass 3: weighted scatter
// One wave32 per edge; each lane handles V = CHTOT/32 consecutive channels (float4 loads for V=4, float2 for V=2).
template <int HEADS, int CHTOT>
__global__ void edge_agg_kernel(const long long* __restrict__ ei, int E, int Etot,
                                const float* __restrict__ hfeat,
                                const float* __restrict__ als,
                                const float* __restrict__ ald,
                                const float* __restrict__ mx,
                                const float* __restrict__ den,
                                float* __restrict__ agg) {
    constexpr int V = CHTOT / 32;
    int gtid = blockIdx.x * blockDim.x + threadIdx.x;
    int eid  = gtid >> 5;
    int lane = gtid & 31;
    if (eid >= Etot) return;
    int s, d;
    edge_endpoints(ei, E, eid, s, d);

    const int c0   = lane * V;
    const int head = c0 / (CHTOT / HEADS);  // all V channels of a lane share one head
    float e = als[s * HEADS + head] + ald[d * HEADS + head];
    e = (e >= 0.f) ? e : 0.2f * e;
    float alpha = __expf(e - mx[d * HEADS + head]) / den[d * HEADS + head];

    const float* hp = hfeat + (size_t)s * CHTOT + c0;
    float vv[V];
    if constexpr (V == 4) {
        const float4 t = *(const float4*)hp;
        vv[0] = t.x; vv[1] = t.y; vv[2] = t.z; vv[3] = t.w;
    } else {
        const float2 t = *(const float2*)hp;
        vv[0] = t.x; vv[1] = t.y;
    }
    float* ap = agg + (size_t)d * CHTOT + c0;
    #pragma unroll
    for (int v = 0; v < V; ++v) atomicAdd(&ap[v], vv[v] * alpha);
}

// ---------------------------------------------------------------- ELU (in place)
__global__ void elu_kernel(float* __restrict__ p, int n) {
    int i = blockIdx.x * blockDim.x + threadIdx.x;
    if (i < n) {
        float v = p[i];
        p[i] = (v > 0.f) ? v : expm1f(v);
    }
}

// ---------------------------------------------------------------- launcher
extern "C" void kernel_launch(void* const* d_in, const int* in_sizes, int n_in,
                              void* d_out, int out_size, void* d_ws, size_t ws_size,
                              hipStream_t stream) {
    const float*     x   = (const float*)d_in[0];
    const long long* ei  = (const long long*)d_in[1];  // int64 edge_index [2,E]
    const float*     W1  = (const float*)d_in[2];
    const float*     a1s = (const float*)d_in[3];
    const float*     a1d = (const float*)d_in[4];
    const float*     b1  = (const float*)d_in[5];
    const float*     W2  = (const float*)d_in[6];
    const float*     a2s = (const float*)d_in[7];
    const float*     a2d = (const float*)d_in[8];
    const float*     b2  = (const float*)d_in[9];
    float*           out = (float*)d_out;

    const int N    = in_sizes[0] / 128;
    const int E    = in_sizes[1] / 2;
    const int Etot = E + N;  // reference appends self-loops

    // workspace layout (floats)
    float* ws   = (float*)d_ws;
    float* h1   = ws;                       // N*128 projected features, layer 1
    float* agg1 = h1 + (size_t)N * 128;     // N*128 aggregation (seeded with b1)
    float* al1s = agg1 + (size_t)N * 128;   // N*4
    float* al1d = al1s + (size_t)N * 4;     // N*4
    float* m1   = al1d + (size_t)N * 4;     // N*4
    float* den1 = m1 + (size_t)N * 4;       // N*4
    float* h2   = den1 + (size_t)N * 4;     // N*64
    float* al2s = h2 + (size_t)N * 64;      // N
    float* al2d = al2s + (size_t)N;         // N
    float* m2   = al2d + (size_t)N;         // N
    float* den2 = m2 + (size_t)N;           // N

    const int T = 256;
    auto blocks = [](long long n, int t) { return (int)((n + t - 1) / t); };

    // ---- init (every call: re-seed accumulators -> deterministic w.r.t. replays)
    fill_f32_kernel<<<blocks((long long)N * 4, T), T, 0, stream>>>(m1, NEG_INF, N * 4);
    fill_f32_kernel<<<blocks((long long)N * 4, T), T, 0, stream>>>(den1, 0.f, N * 4);
    fill_f32_kernel<<<blocks(N, T), T, 0, stream>>>(m2, NEG_INF, N);
    fill_f32_kernel<<<blocks(N, T), T, 0, stream>>>(den2, 0.f, N);
    fill_bias_kernel<<<blocks((long long)N * 128, T), T, 0, stream>>>(agg1, b1, N, 128);
    fill_bias_kernel<<<blocks((long long)N * 64, T), T, 0, stream>>>(out, b2, N, 64);

    // ---- layer 1: h1 = x @ W1 (WMMA), attention, segment softmax, scatter, +b1, ELU
    gemm_wmma_kernel<128, 128><<<(N + 15) / 16, 256, 0, stream>>>(x, W1, h1, N);
    att_logits_kernel<4, 32><<<blocks((long long)N * 4, T), T, 0, stream>>>(
        h1, a1s, a1d, al1s, al1d, N);
    edge_max_kernel<4><<<blocks((long long)Etot * 4, T), T, 0, stream>>>(
        ei, E, Etot, al1s, al1d, m1);
    edge_sum_kernel<4><<<blocks((long long)Etot * 4, T), T, 0, stream>>>(
        ei, E, Etot, al1s, al1d, m1, den1);
    edge_agg_kernel<4, 128><<<blocks((long long)Etot * 32, T), T, 0, stream>>>(
        ei, E, Etot, h1, al1s, al1d, m1, den1, agg1);
    elu_kernel<<<blocks((long long)N * 128, T), T, 0, stream>>>(agg1, N * 128);

    // ---- layer 2: h2 = elu_out @ W2 (WMMA), 1 head x 64 ch, scatter into d_out (+b2 seeded)
    gemm_wmma_kernel<128, 64><<<(N + 15) / 16, 128, 0, stream>>>(agg1, W2, h2, N);
    att_logits_kernel<1, 64><<<blocks(N, T), T, 0, stream>>>(h2, a2s, a2d, al2s, al2d, N);
    edge_max_kernel<1><<<blocks(Etot, T), T, 0, stream>>>(ei, E, Etot, al2s, al2d, m2);
    edge_sum_kernel<1><<<blocks(Etot, T), T, 0, stream>>>(ei, E, Etot, al2s, al2d, m2, den2);
    edge_agg_kernel<1, 64><<<blocks((long long)Etot * 32, T), T, 0, stream>>>(
        ei, E, Etot, h2, al2s, al2d, m2, den2, out);
}